// GCNNet_71614284694111
// MI455X (gfx1250) — compile-verified
//
#include <hip/hip_runtime.h>

#define NN 50000
#define NE 800000
#define DD 64
#define NL 3
#define NG 512

typedef float v2f __attribute__((ext_vector_type(2)));
typedef float v8f __attribute__((ext_vector_type(8)));

// ---------------------------------------------------------------------------
// Degree / dinv
// ---------------------------------------------------------------------------
__global__ __launch_bounds__(256) void zero_u32(unsigned int* p, int n) {
  int i = blockIdx.x * blockDim.x + threadIdx.x;
  if (i < n) p[i] = 0u;
}

__global__ __launch_bounds__(256) void zero_f32(float* p, int n) {
  int i = blockIdx.x * blockDim.x + threadIdx.x;
  if (i < n) p[i] = 0.0f;
}

__global__ __launch_bounds__(256) void edge_count(const int* __restrict__ dst,
                                                  unsigned int* __restrict__ cnt,
                                                  int ne) {
  int e = blockIdx.x * blockDim.x + threadIdx.x;
  if (e < ne) atomicAdd(&cnt[dst[e]], 1u);
}

__global__ __launch_bounds__(256) void make_dinv(const unsigned int* __restrict__ cnt,
                                                 float* __restrict__ dinv, int n) {
  int i = blockIdx.x * blockDim.x + threadIdx.x;
  if (i < n) dinv[i] = rsqrtf((float)cnt[i] + 1.0f);
}

// ---------------------------------------------------------------------------
// Dense GEMM  Out[N,64] = H[N,64] @ W[64,64]  via V_WMMA_F32_16X16X4_F32.
// One wave per 16-row tile; 4 x v8f accumulators cover all 64 output cols.
// ---------------------------------------------------------------------------
__global__ __launch_bounds__(256)
void gemm_hW(const float* __restrict__ H, const float* __restrict__ W,
             float* __restrict__ Out, int n_tiles) {
  int wave = (int)((blockIdx.x * blockDim.x + threadIdx.x) >> 5);
  int lane = threadIdx.x & 31;
  if (wave >= n_tiles) return;  // uniform per wave -> EXEC stays all-ones

  const int row0 = wave * 16;
  const int m  = lane & 15;   // row (A) / col (B,D) within tile
  const int kh = lane >> 4;   // lane-half selects K pair {0,1} vs {2,3}

  v8f acc0 = {}, acc1 = {}, acc2 = {}, acc3 = {};
  const float* hrow = H + (size_t)(row0 + m) * DD;

  for (int kk = 0; kk < DD; kk += 4) {
    const int ka = kk + kh * 2;
    v2f a;
    a.x = hrow[ka];
    a.y = hrow[ka + 1];

    const float* w0 = W + (size_t)ka * DD;
    const float* w1 = w0 + DD;
    v2f b0, b1, b2, b3;
    b0.x = w0[m];      b0.y = w1[m];
    b1.x = w0[m + 16]; b1.y = w1[m + 16];
    b2.x = w0[m + 32]; b2.y = w1[m + 32];
    b3.x = w0[m + 48]; b3.y = w1[m + 48];

    acc0 = __builtin_amdgcn_wmma_f32_16x16x4_f32(false, a, false, b0, (short)0, acc0, false, false);
    acc1 = __builtin_amdgcn_wmma_f32_16x16x4_f32(false, a, false, b1, (short)0, acc1, false, false);
    acc2 = __builtin_amdgcn_wmma_f32_16x16x4_f32(false, a, false, b2, (short)0, acc2, false, false);
    acc3 = __builtin_amdgcn_wmma_f32_16x16x4_f32(false, a, false, b3, (short)0, acc3, false, false);
  }

  // D layout: lane L, VGPR v -> row = v + 8*(L/16), col = L%16
  const int rbase = row0 + (kh << 3);
#pragma unroll
  for (int v = 0; v < 8; ++v) {
    float* orow = Out + (size_t)(rbase + v) * DD + m;
    orow[0]  = acc0[v];
    orow[16] = acc1[v];
    orow[32] = acc2[v];
    orow[48] = acc3[v];
  }
}

// ---------------------------------------------------------------------------
// agg[i] = hW[i] * dinv[i]^2 + b   (self-loop contribution + bias)
// 16 threads per node, float4 each.
// ---------------------------------------------------------------------------
__global__ __launch_bounds__(256)
void agg_init(const float* __restrict__ hW, const float* __restrict__ dinv,
              const float* __restrict__ b, float* __restrict__ agg, int n_nodes) {
  int gid = blockIdx.x * blockDim.x + threadIdx.x;
  int i = gid >> 4;
  int c = gid & 15;
  if (i >= n_nodes) return;
  float di = dinv[i];
  float s = di * di;
  float4 v  = ((const float4*)(hW + (size_t)i * DD))[c];
  float4 bb = ((const float4*)b)[c];
  float4 o;
  o.x = v.x * s + bb.x;
  o.y = v.y * s + bb.y;
  o.z = v.z * s + bb.z;
  o.w = v.w * s + bb.w;
  ((float4*)(agg + (size_t)i * DD))[c] = o;
}

// ---------------------------------------------------------------------------
// Edge scatter: agg[dst] += hW[src] * dinv[src]*dinv[dst].
// 16 threads per edge, float4 gather + 4 f32 atomics. hW/agg are 12.8MB each
// -> entire working set resident in the 192MB L2.
// ---------------------------------------------------------------------------
__global__ __launch_bounds__(256)
void edge_agg(const float* __restrict__ hW, const float* __restrict__ dinv,
              const int* __restrict__ src, const int* __restrict__ dst,
              float* __restrict__ agg, int ne) {
  long long gid = (long long)blockIdx.x * blockDim.x + threadIdx.x;
  int e = (int)(gid >> 4);
  int c = (int)(gid & 15);
  if (e >= ne) return;
  int s = src[e];
  int t = dst[e];
  float norm = dinv[s] * dinv[t];
  float4 v = ((const float4*)(hW + (size_t)s * DD))[c];
  float* out = agg + (size_t)t * DD + c * 4;
  atomicAdd(out + 0, v.x * norm);
  atomicAdd(out + 1, v.y * norm);
  atomicAdd(out + 2, v.z * norm);
  atomicAdd(out + 3, v.w * norm);
}

// ---------------------------------------------------------------------------
// Pooling
// ---------------------------------------------------------------------------
__global__ __launch_bounds__(256)
void pool_sum(const float* __restrict__ h, const int* __restrict__ batch,
              float* __restrict__ psum, int n_nodes) {
  int gid = blockIdx.x * blockDim.x + threadIdx.x;
  int i = gid >> 4;
  int c = gid & 15;
  if (i >= n_nodes) return;
  int g = batch[i];
  float4 v = ((const float4*)(h + (size_t)i * DD))[c];
  float* out = psum + (size_t)g * DD + c * 4;
  atomicAdd(out + 0, v.x);
  atomicAdd(out + 1, v.y);
  atomicAdd(out + 2, v.z);
  atomicAdd(out + 3, v.w);
}

__global__ __launch_bounds__(256)
void pool_cnt(const int* __restrict__ batch, float* __restrict__ pcnt, int n_nodes) {
  int i = blockIdx.x * blockDim.x + threadIdx.x;
  if (i < n_nodes) atomicAdd(&pcnt[batch[i]], 1.0f);
}

__global__ __launch_bounds__(256)
void pool_final(const float* __restrict__ psum, const float* __restrict__ pcnt,
                float* __restrict__ out) {
  int gid = blockIdx.x * blockDim.x + threadIdx.x;  // NG*DD threads
  if (gid >= NG * DD) return;
  int g = gid / DD;
  out[gid] = psum[gid] / fmaxf(pcnt[g], 1.0f);
}

// ---------------------------------------------------------------------------
// Launch
// ---------------------------------------------------------------------------
extern "C" void kernel_launch(void* const* d_in, const int* in_sizes, int n_in,
                              void* d_out, int out_size, void* d_ws, size_t ws_size,
                              hipStream_t stream) {
  const float* x    = (const float*)d_in[0];             // [NN, DD]
  const float* Ws   = (const float*)d_in[1];             // [NL, DD, DD]
  const float* bs   = (const float*)d_in[2];             // [NL, DD]
  const int*   eidx = (const int*)d_in[3];               // [2, NE]
  const int*   batch = (const int*)d_in[4];              // [NN]
  const int* src = eidx;
  const int* dst = eidx + NE;
  float* out = (float*)d_out;

  // Workspace layout (bytes)
  char* ws = (char*)d_ws;
  const size_t nodeBuf = (size_t)NN * DD * sizeof(float);  // 12.8 MB, 256B aligned
  float*        hW   = (float*)(ws);
  float*        hA   = (float*)(ws + nodeBuf);
  float*        hB   = (float*)(ws + 2 * nodeBuf);
  float*        dinv = (float*)(ws + 3 * nodeBuf);
  unsigned int* cnt  = (unsigned int*)(ws + 3 * nodeBuf + (size_t)NN * 4);
  float*        psum = (float*)(ws + 3 * nodeBuf + (size_t)NN * 8);
  float*        pcnt = psum + (size_t)NG * DD;             // contiguous after psum

  const int T = 256;
  const int gNN   = (NN + T - 1) / T;
  const int gNE   = (NE + T - 1) / T;
  const int gNN16 = (NN * 16 + T - 1) / T;
  const int gNE16 = (NE * 16 + T - 1) / T;

  // 1) degrees -> dinv
  zero_u32<<<gNN, T, 0, stream>>>(cnt, NN);
  edge_count<<<gNE, T, 0, stream>>>(dst, cnt, NE);
  make_dinv<<<gNN, T, 0, stream>>>(cnt, dinv, NN);

  // 2) three GCN layers, ping-pong hA/hB
  const int n_tiles = NN / 16;                         // 3125, exact
  const int gGemm = (n_tiles * 32 + T - 1) / T;        // waves -> threads
  const float* hin = x;
  float* hout = hA;
  float* hother = hB;
  for (int l = 0; l < NL; ++l) {
    const float* W = Ws + (size_t)l * DD * DD;
    const float* b = bs + (size_t)l * DD;
    gemm_hW<<<gGemm, T, 0, stream>>>(hin, W, hW, n_tiles);
    agg_init<<<gNN16, T, 0, stream>>>(hW, dinv, b, hout, NN);
    edge_agg<<<gNE16, T, 0, stream>>>(hW, dinv, src, dst, hout, NE);
    hin = hout;
    float* tmp = hout == hA ? hB : hA;
    hother = hout;
    hout = tmp;
    (void)hother;
  }

  // 3) global mean pool
  zero_f32<<<(NG * DD + NG + T - 1) / T, T, 0, stream>>>(psum, NG * DD + NG);
  pool_sum<<<gNN16, T, 0, stream>>>(hin, batch, psum, NN);
  pool_cnt<<<gNN, T, 0, stream>>>(batch, pcnt, NN);
  pool_final<<<(NG * DD + T - 1) / T, T, 0, stream>>>(psum, pcnt, out);

  (void)in_sizes; (void)n_in; (void)out_size; (void)ws_size;
}